// Permute_67001489817758
// MI455X (gfx1250) — compile-verified
//
#include <hip/hip_runtime.h>

typedef float v4f __attribute__((ext_vector_type(4)));

#define NUM_P 16
#define STEP 4
#define NUM_BLOCKS 256
#define DIM 1024
#define BATCH 4096

// ---------------------------------------------------------------------------
// Kernel 1: recover the block permutation hidden in the dense T matrices.
// T[p] is [D,D]; block (r,c) is a 4x4 identity iff perm_p[r]==c, so it is
// enough to probe element T[p][4r][4c]. One workgroup per (p,r) pair; thread
// c probes column-block c; exactly one thread writes (values are exact 0/1).
// ---------------------------------------------------------------------------
__global__ __launch_bounds__(256) void extract_perm_kernel(
    const float* __restrict__ T, int* __restrict__ perm) {
  const int pr = blockIdx.x;                 // p*256 + r
  const int r  = pr & (NUM_BLOCKS - 1);
  const int p  = pr >> 8;
  const int c  = threadIdx.x;                // candidate column block
  const float v =
      T[(size_t)p * DIM * DIM + (size_t)(STEP * r) * DIM + (size_t)STEP * c];
  if (v != 0.0f) perm[pr] = c;
}

// ---------------------------------------------------------------------------
// Kernel 2: out[(i*B + b), 4r+j] = x[b, 4*perm[indices[i]][r] + j].
// One workgroup per batch row b:
//   - async-copy the 4 KB row x[b,:] into LDS (256 lanes x B128, ASYNCcnt)
//   - each of the 8 waves emits 2 of the 16 permuted output rows, gathering
//     16B blocks from LDS (ds_load_b128) and streaming them out with
//     non-temporal b128 stores (output is write-once, keep it out of L2).
// ---------------------------------------------------------------------------
__global__ __launch_bounds__(256) void permute_gather_kernel(
    const float* __restrict__ x, const int* __restrict__ indices,
    const int* __restrict__ perm, float* __restrict__ out) {
  __shared__ v4f lds_row[NUM_BLOCKS];        // 4 KB staging for one x row
  const int b    = blockIdx.x;
  const int tid  = threadIdx.x;
  const int lane = tid & 31;
  const int wave = tid >> 5;

  // Async global->LDS copy of x[b, :]. Generic pointers to LDS carry the
  // DS-relative offset in their low 32 bits (ISA 10.2 aperture mapping).
  {
    const unsigned lds_off = (unsigned)(size_t)(&lds_row[0]) + (unsigned)tid * 16u;
    const unsigned goff    = (unsigned)tid * 16u;
    const unsigned long long base =
        (unsigned long long)(x + (size_t)b * DIM);
    asm volatile("global_load_async_to_lds_b128 %0, %1, %2"
                 :: "v"(lds_off), "v"(goff), "s"(base)
                 : "memory");
    asm volatile("s_wait_asynccnt 0" ::: "memory");
  }
  __syncthreads();

  // Each wave produces output groups i = 2*wave and 2*wave+1.
#pragma unroll
  for (int g = 0; g < 2; ++g) {
    const int i = wave * 2 + g;              // output row-group 0..15
    const int p = indices[i];                // uniform per wave -> scalar load
    const int* __restrict__ pp = perm + p * NUM_BLOCKS;
    float* __restrict__ orow = out + ((size_t)i * BATCH + b) * DIM;
#pragma unroll
    for (int it = 0; it < 8; ++it) {
      const int r = it * 32 + lane;          // output block index
      const int c = pp[r];                   // source block index
      const v4f v = lds_row[c];              // ds_load_b128 gather from LDS
      __builtin_nontemporal_store(v, (v4f*)orow + r);  // b128 NT store
    }
  }
}

extern "C" void kernel_launch(void* const* d_in, const int* in_sizes, int n_in,
                              void* d_out, int out_size, void* d_ws, size_t ws_size,
                              hipStream_t stream) {
  const float* x       = (const float*)d_in[0];   // [4096, 1024] f32
  const float* T       = (const float*)d_in[1];   // [16, 1024, 1024] f32
  const int*   indices = (const int*)d_in[2];     // [16] i32
  float* out = (float*)d_out;                     // [16*4096, 1024] f32
  int* perm = (int*)d_ws;                         // 16*256 ints = 16 KB scratch

  extract_perm_kernel<<<NUM_P * NUM_BLOCKS, NUM_BLOCKS, 0, stream>>>(T, perm);
  permute_gather_kernel<<<BATCH, 256, 0, stream>>>(x, indices, perm, out);
}